// AUGGCN_63539746177183
// MI455X (gfx1250) — compile-verified
//
#include <hip/hip_runtime.h>
#include <hip/hip_bf16.h>
#include <math.h>

typedef __attribute__((ext_vector_type(2))) float v2f;
typedef __attribute__((ext_vector_type(8))) float v8f;

#define F0 165   // input features
#define F1 128   // hidden features
#define K_PAD 168        // 165 rounded up to multiple of 4
#define AS_STRIDE 170    // LDS row stride (even, bank-conflict-free for lo*stride%64)

// ---------------- zero fill ----------------
__global__ void k_zero(float* p, long long n) {
    long long i = (long long)blockIdx.x * blockDim.x + threadIdx.x;
    long long stride = (long long)gridDim.x * blockDim.x;
    for (; i < n; i += stride) p[i] = 0.0f;
}

// ---------------- degree over edges + self loops ----------------
__global__ void k_degree(const int* __restrict__ eidx, float* __restrict__ deg,
                         int E, int N) {
    int e = blockIdx.x * blockDim.x + threadIdx.x;
    int total = E + N;
    if (e >= total) return;
    int dst = (e < E) ? eidx[E + e] : (e - E);
    atomicAdd(&deg[dst], 1.0f);
}

// ---------------- deg -> deg^{-1/2} (in place) ----------------
__global__ void k_dinv(float* __restrict__ deg, int N) {
    int i = blockIdx.x * blockDim.x + threadIdx.x;
    if (i >= N) return;
    float d = deg[i];
    deg[i] = (d > 0.0f) ? rsqrtf(d) : 0.0f;
}

// ---------------- xw = x @ W1 via V_WMMA_F32_16X16X4_F32 ----------------
// grid.x = N/16 blocks, 256 threads = 8 waves. Wave w computes the 16x16
// output tile covering columns [16w,16w+16). A-tile (16 rows x K_PAD) staged
// in LDS and shared by all 8 waves.
__global__ void k_gemm1_wmma(const float* __restrict__ x,
                             const float* __restrict__ W1,
                             float* __restrict__ xw, int N) {
    __shared__ float As[16 * AS_STRIDE];

    const int tid  = threadIdx.x;
    const int row0 = blockIdx.x * 16;

    // cooperative load of A tile (zero-pad cols >= F0)
    for (int idx = tid; idx < 16 * AS_STRIDE; idx += 256) {
        int r = idx / AS_STRIDE;
        int c = idx - r * AS_STRIDE;
        As[idx] = (c < F0) ? x[(long long)(row0 + r) * F0 + c] : 0.0f;
    }
    __syncthreads();

    const int wid  = tid >> 5;          // wave id 0..7 -> column tile
    const int lane = tid & 31;
    const int lo   = lane & 15;         // row (A) / col (B,C,D) within tile
    const int kb   = (lane >> 4) << 1;  // K sub-offset: 0 for lanes 0-15, 2 for 16-31
    const int col  = (wid << 4) + lo;   // global output column 0..127

    v8f acc = {};
    for (int k = 0; k < K_PAD; k += 4) {
        const int kk = k + kb;
        v2f a;
        a.x = As[lo * AS_STRIDE + kk];
        a.y = As[lo * AS_STRIDE + kk + 1];

        // B rows kk, kk+1 of W1 (clamp index to stay in-bounds, select 0 for pad)
        int k0 = kk;     int k0c = (k0 < F0) ? k0 : (F0 - 1);
        int k1 = kk + 1; int k1c = (k1 < F0) ? k1 : (F0 - 1);
        float b0 = W1[k0c * F1 + col];
        float b1 = W1[k1c * F1 + col];
        v2f b;
        b.x = (k0 < F0) ? b0 : 0.0f;
        b.y = (k1 < F0) ? b1 : 0.0f;

        acc = __builtin_amdgcn_wmma_f32_16x16x4_f32(
            false, a, false, b, (short)0, acc, false, false);
    }

    // store D: VGPR i -> row (i + 8*(lane>=16)), col = lo
    const int mhi = (lane >> 4) << 3;
    #pragma unroll
    for (int i = 0; i < 8; ++i) {
        int m = i + mhi;
        xw[(long long)(row0 + m) * F1 + col] = acc[i];
    }
}

// ---------------- layer-1 aggregation: h1[dst] += xw[src]*norm ----------------
// 256 threads = 2 edges x 128 features; fully coalesced per edge.
__global__ void k_agg1(const int* __restrict__ eidx,
                       const float* __restrict__ dinv,
                       const float* __restrict__ xw,
                       float* __restrict__ h1, int E, int N) {
    int le   = threadIdx.x >> 7;          // 0..1
    int feat = threadIdx.x & 127;
    int e = blockIdx.x * 2 + le;
    int total = E + N;
    if (e >= total) return;
    int src = (e < E) ? eidx[e]     : (e - E);
    int dst = (e < E) ? eidx[E + e] : (e - E);
    float nrm = dinv[src] * dinv[dst];
    float m = xw[(long long)src * F1 + feat] * nrm;
    atomicAdd(&h1[(long long)dst * F1 + feat], m);
}

// ---------------- h1 = relu(acc+b1); xw2 = h1 @ W2 (128->2), wave per node ----
__global__ void k_layer2pre(const float* __restrict__ h1acc,
                            const float* __restrict__ b1,
                            const float* __restrict__ W2,
                            float* __restrict__ xw2, int N) {
    int gtid = blockIdx.x * blockDim.x + threadIdx.x;
    int node = gtid >> 5;
    int lane = gtid & 31;
    if (node >= N) return;
    float s0 = 0.0f, s1 = 0.0f;
    for (int f = lane; f < F1; f += 32) {
        float h = h1acc[(long long)node * F1 + f] + b1[f];
        h = fmaxf(h, 0.0f);
        s0 += h * W2[f * 2 + 0];
        s1 += h * W2[f * 2 + 1];
    }
    #pragma unroll
    for (int off = 16; off > 0; off >>= 1) {
        s0 += __shfl_down(s0, off, 32);
        s1 += __shfl_down(s1, off, 32);
    }
    if (lane == 0) {
        xw2[node * 2 + 0] = s0;
        xw2[node * 2 + 1] = s1;
    }
}

// ---------------- layer-2 aggregation (2 features), thread per edge ----------
__global__ void k_agg2(const int* __restrict__ eidx,
                       const float* __restrict__ dinv,
                       const float* __restrict__ xw2,
                       float* __restrict__ h2, int E, int N) {
    int e = blockIdx.x * blockDim.x + threadIdx.x;
    int total = E + N;
    if (e >= total) return;
    int src = (e < E) ? eidx[e]     : (e - E);
    int dst = (e < E) ? eidx[E + e] : (e - E);
    float nrm = dinv[src] * dinv[dst];
    atomicAdd(&h2[dst * 2 + 0], xw2[src * 2 + 0] * nrm);
    atomicAdd(&h2[dst * 2 + 1], xw2[src * 2 + 1] * nrm);
}

// ---------------- relu + classifier head + sigmoid ----------------
__global__ void k_final(const float* __restrict__ h2,
                        const float* __restrict__ b2,
                        const float* __restrict__ Wc,
                        const float* __restrict__ bc,
                        float* __restrict__ out, int N) {
    int i = blockIdx.x * blockDim.x + threadIdx.x;
    if (i >= N) return;
    float a = fmaxf(h2[i * 2 + 0] + b2[0], 0.0f);
    float b = fmaxf(h2[i * 2 + 1] + b2[1], 0.0f);
    float z = a * Wc[0] + b * Wc[1] + bc[0];
    out[i] = 1.0f / (1.0f + expf(-z));
}

extern "C" void kernel_launch(void* const* d_in, const int* in_sizes, int n_in,
                              void* d_out, int out_size, void* d_ws, size_t ws_size,
                              hipStream_t stream) {
    const float* x    = (const float*)d_in[0];
    const int*   eidx = (const int*)  d_in[1];   // [2,E] row-major: src then dst
    const float* W1   = (const float*)d_in[2];
    const float* b1   = (const float*)d_in[3];
    const float* W2   = (const float*)d_in[4];
    const float* b2   = (const float*)d_in[5];
    const float* Wc   = (const float*)d_in[6];
    const float* bc   = (const float*)d_in[7];
    float* out = (float*)d_out;

    const int N = in_sizes[0] / F0;     // 100000
    const int E = in_sizes[1] / 2;      // 1600000
    const int total = E + N;

    // workspace layout (floats)
    float* ws    = (float*)d_ws;
    float* dinv  = ws;                               // N      (deg, then deg^-1/2)
    float* xw    = dinv + N;                         // N*128
    float* h1    = xw   + (long long)N * F1;         // N*128
    float* xw2   = h1   + (long long)N * F1;         // N*2
    float* h2    = xw2  + (long long)N * 2;          // N*2

    // zero the accumulators (atomics targets) every call
    k_zero<<<2048, 256, 0, stream>>>(dinv, N);
    k_zero<<<2048, 256, 0, stream>>>(h1, (long long)N * F1);
    k_zero<<<2048, 256, 0, stream>>>(h2, (long long)N * 2);

    k_degree<<<(total + 255) / 256, 256, 0, stream>>>(eidx, dinv, E, N);
    k_dinv  <<<(N + 255) / 256, 256, 0, stream>>>(dinv, N);

    k_gemm1_wmma<<<N / 16, 256, 0, stream>>>(x, W1, xw, N);

    k_agg1<<<(total + 1) / 2, 256, 0, stream>>>(eidx, dinv, xw, h1, E, N);

    k_layer2pre<<<(N * 32 + 255) / 256, 256, 0, stream>>>(h1, b1, W2, xw2, N);

    k_agg2<<<(total + 255) / 256, 256, 0, stream>>>(eidx, dinv, xw2, h2, E, N);

    k_final<<<(N + 255) / 256, 256, 0, stream>>>(h2, b2, Wc, bc, out, N);
}